// GraphSAGEModel_43576738185797
// MI455X (gfx1250) — compile-verified
//
#include <hip/hip_runtime.h>

typedef __attribute__((ext_vector_type(2))) float v2f;
typedef __attribute__((ext_vector_type(8))) float v8f;

// ---------------------------------------------------------------------------
// zero-fill (graph-capture-safe replacement for memset)
// ---------------------------------------------------------------------------
__global__ void __launch_bounds__(256) zero_f32_kernel(float* __restrict__ p, long long n) {
  long long i = (long long)blockIdx.x * blockDim.x + threadIdx.x;
  if (i < n) p[i] = 0.0f;
}

// ---------------------------------------------------------------------------
// Edge scatter-add: one wave per edge. D=128 -> float4/lane, D=64 -> float2/lane.
// msg buffers fit in the 192MB L2, so these atomics resolve in L2.
// ---------------------------------------------------------------------------
template <int D>
__global__ void __launch_bounds__(256) scatter_add_kernel(
    const float* __restrict__ x,
    const long long* __restrict__ src,
    const long long* __restrict__ dst,
    float* __restrict__ msg,
    float* __restrict__ deg,   // nullptr => skip degree accumulation
    long long E) {
  long long gid = (long long)blockIdx.x * blockDim.x + threadIdx.x;
  long long e = gid >> 5;
  int lane = threadIdx.x & 31;
  if (e >= E) return;
  long long s = src[e];
  long long d = dst[e];
  if (D == 128) {
    const float4 v = ((const float4*)(x + s * 128))[lane];
    float* m = msg + d * 128 + lane * 4;
    atomicAdd(m + 0, v.x);
    atomicAdd(m + 1, v.y);
    atomicAdd(m + 2, v.z);
    atomicAdd(m + 3, v.w);
  } else {
    const float2 v = ((const float2*)(x + s * 64))[lane];
    float* m = msg + d * 64 + lane * 2;
    atomicAdd(m + 0, v.x);
    atomicAdd(m + 1, v.y);
  }
  if (deg != nullptr && lane == 0) atomicAdd(deg + d, 1.0f);
}

// ---------------------------------------------------------------------------
// inv[i] = 1 / max(deg[i], 1)
// ---------------------------------------------------------------------------
__global__ void __launch_bounds__(256) invdeg_kernel(
    const float* __restrict__ deg, float* __restrict__ inv, int n) {
  int i = blockIdx.x * blockDim.x + threadIdx.x;
  if (i < n) inv[i] = 1.0f / fmaxf(deg[i], 1.0f);
}

// ---------------------------------------------------------------------------
// Fused SAGEConv GEMM on the WMMA pipe (fp32, full reference precision):
//   out[m, n] = relu( (msg[m,:] * invdeg[m]) @ Wl[n,:] + bias[n] + root[m,:] @ Wr[n,:] )
// Block = 16 node rows; 4 waves, wave w owns columns [16w, 16w+16).  H == 64.
// V_WMMA_F32_16X16X4_F32 layouts (32-bit A 16x4 / B 4x16):
//   lane half 0 (lanes 0-15):  K = k+0 (vgpr0), k+1 (vgpr1)
//   lane half 1 (lanes 16-31): K = k+2 (vgpr0), k+3 (vgpr1)
// C/D: vgpr v, lanes 0-15 -> row m0+v, lanes 16-31 -> row m0+v+8, col n0+(lane&15)
// ---------------------------------------------------------------------------
__global__ void __launch_bounds__(128) sage_gemm_wmma_kernel(
    const float* __restrict__ Amsg,   // [N,K] neighbor sums (scaled by invdeg on load)
    const float* __restrict__ invdeg, // [N]
    const float* __restrict__ Aroot,  // [N,K] root features
    const float* __restrict__ Wl,     // [64,K] row-major
    const float* __restrict__ Wr,     // [64,K] row-major
    const float* __restrict__ bias,   // [64]
    float* __restrict__ out,          // [N,64]
    int N, int K) {
  const int m0 = blockIdx.x * 16;
  const int n0 = (threadIdx.x >> 5) * 16;   // wave id * 16
  const int lane = threadIdx.x & 31;
  const int l15 = lane & 15;
  const int hi = lane >> 4;                 // lane half (selects K pair)

  int mA = m0 + l15;
  if (mA >= N) mA = N - 1;                  // clamp loads for tail tile (uniform EXEC kept)
  const float sA = invdeg[mA];

  const float* a1 = Amsg  + (long long)mA * K + 2 * hi;
  const float* a2 = Aroot + (long long)mA * K + 2 * hi;
  const float* b1 = Wl + (long long)(n0 + l15) * K + 2 * hi;
  const float* b2 = Wr + (long long)(n0 + l15) * K + 2 * hi;

  v8f c = {};
  // neighbor-mean path: (msg * invdeg) @ Wl^T
  for (int k = 0; k < K; k += 4) {
    v2f a, b;
    a[0] = a1[k] * sA; a[1] = a1[k + 1] * sA;
    b[0] = b1[k];      b[1] = b1[k + 1];
    c = __builtin_amdgcn_wmma_f32_16x16x4_f32(false, a, false, b, (short)0, c, false, false);
  }
  // root path: x @ Wr^T (accumulates into same tile)
  for (int k = 0; k < K; k += 4) {
    v2f a, b;
    a[0] = a2[k];  a[1] = a2[k + 1];
    b[0] = b2[k];  b[1] = b2[k + 1];
    c = __builtin_amdgcn_wmma_f32_16x16x4_f32(false, a, false, b, (short)0, c, false, false);
  }

  const int col = n0 + l15;
  const float bv = bias[col];
#pragma unroll
  for (int v = 0; v < 8; ++v) {
    const int r = m0 + v + 8 * hi;
    if (r < N) {
      float val = fmaxf(c[v] + bv, 0.0f);   // bias + ReLU fused
      out[(long long)r * 64 + col] = val;
    }
  }
}

// ---------------------------------------------------------------------------
// Final head: out[i] = dot(h2[i, 0:64], Wc) + bc.  One wave per node.
// ---------------------------------------------------------------------------
__global__ void __launch_bounds__(256) classifier_kernel(
    const float* __restrict__ h, const float* __restrict__ Wc,
    const float* __restrict__ bc, float* __restrict__ out, int N) {
  int w = blockIdx.x * (blockDim.x >> 5) + (threadIdx.x >> 5);
  int lane = threadIdx.x & 31;
  if (w >= N) return;
  float2 a = ((const float2*)(h + (long long)w * 64))[lane];
  float2 b = ((const float2*)Wc)[lane];
  float s = a.x * b.x + a.y * b.y;
#pragma unroll
  for (int off = 16; off > 0; off >>= 1) s += __shfl_down(s, off, 32);
  if (lane == 0) out[w] = s + bc[0];
}

// ---------------------------------------------------------------------------
// Orchestration
// ---------------------------------------------------------------------------
extern "C" void kernel_launch(void* const* d_in, const int* in_sizes, int n_in,
                              void* d_out, int out_size, void* d_ws, size_t ws_size,
                              hipStream_t stream) {
  const int IN = 128, H = 64;
  const float*     x    = (const float*)d_in[0];
  const long long* ei   = (const long long*)d_in[1];
  const float*     Wl1  = (const float*)d_in[2];
  const float*     bl1  = (const float*)d_in[3];
  const float*     Wr1  = (const float*)d_in[4];
  const float*     Wl2  = (const float*)d_in[5];
  const float*     bl2  = (const float*)d_in[6];
  const float*     Wr2  = (const float*)d_in[7];
  const float*     Wc   = (const float*)d_in[8];
  const float*     bc   = (const float*)d_in[9];

  const int       N = in_sizes[0] / IN;
  const long long E = in_sizes[1] / 2;
  const long long* src = ei;
  const long long* dst = ei + E;

  // Workspace carve (floats). msg1 region [N*128] is reused: msg2 = first
  // N*64, h2 = second N*64 (GEMM2 reads msg2, writes h2 -> disjoint halves).
  float* msg1 = (float*)d_ws;
  float* deg  = msg1 + (long long)N * IN;
  float* inv  = deg + N;
  float* h1   = inv + N;
  float* msg2 = msg1;
  float* h2   = msg1 + (long long)N * H;

  // ---- Layer 1 ----
  {
    long long n = (long long)N * IN + N;            // msg1 + deg (contiguous)
    zero_f32_kernel<<<(unsigned)((n + 255) / 256), 256, 0, stream>>>(msg1, n);
  }
  {
    long long thr = E * 32;
    scatter_add_kernel<128><<<(unsigned)((thr + 255) / 256), 256, 0, stream>>>(
        x, src, dst, msg1, deg, E);
  }
  invdeg_kernel<<<(N + 255) / 256, 256, 0, stream>>>(deg, inv, N);
  sage_gemm_wmma_kernel<<<(N + 15) / 16, 128, 0, stream>>>(
      msg1, inv, x, Wl1, Wr1, bl1, h1, N, IN);

  // ---- Layer 2 (same edges -> same degree) ----
  {
    long long n = (long long)N * H;
    zero_f32_kernel<<<(unsigned)((n + 255) / 256), 256, 0, stream>>>(msg2, n);
  }
  {
    long long thr = E * 32;
    scatter_add_kernel<64><<<(unsigned)((thr + 255) / 256), 256, 0, stream>>>(
        h1, src, dst, msg2, (float*)nullptr, E);
  }
  sage_gemm_wmma_kernel<<<(N + 15) / 16, 128, 0, stream>>>(
      msg2, inv, h1, Wl2, Wr2, bl2, h2, N, H);

  // ---- Head ----
  classifier_kernel<<<(N + 7) / 8, 256, 0, stream>>>(h2, Wc, bc, (float*)d_out, N);
}